// Detection_loss_5566277615628
// MI455X (gfx1250) — compile-verified
//
#include <hip/hip_runtime.h>
#include <stdint.h>

typedef float v2f __attribute__((ext_vector_type(2)));
typedef float v8f __attribute__((ext_vector_type(8)));

constexpr int B_ = 16, N_ = 8, D_ = 24, H_ = 48, W_ = 48;
constexpr int A_  = D_ * H_ * W_;   // 55296
constexpr int BA_ = B_ * A_;        // 884736
constexpr int BN_ = B_ * N_;        // 128
constexpr int HW_ = H_ * W_;        // 2304
constexpr int PREP_STRIDE = 20;
// prep per (b,n): [0..2]=ctr_grid [3..5]=half [6..11]=bbox6 [12..14]=lo [15..17]=hi [18]=keep [19]=rej
// accum: [0..15]=pos_sum per image, [16]=cls mean, [17]=shape sum, [18]=off sum, [19]=diou sum, [20]=fg count

__global__ void k_init(int* pos_n, int* ign, float* accum, int* npos) {
  int i = blockIdx.x * blockDim.x + threadIdx.x;
  if (i < BA_) { pos_n[i] = N_; ign[i] = 0; }
  if (i < 24) accum[i] = 0.f;
  if (i < 16) npos[i] = 0;
}

__global__ void k_prep(const float* __restrict__ ann, float* __restrict__ prep) {
  int i = threadIdx.x;
  if (i >= BN_) return;
  const float* a = ann + i * 7;
  float c0 = a[0], c1 = a[1], c2 = a[2];
  float s0 = a[3], s1 = a[4], s2 = a[5];
  float lab = a[6];
  bool has = lab > -1.f;
  float lo0 = fmaxf(c0 - 0.5f * s0, 0.f), lo1 = fmaxf(c1 - 0.5f * s1, 0.f), lo2 = fmaxf(c2 - 0.5f * s2, 0.f);
  float hi0 = fminf(c0 + 0.5f * s0, 96.f), hi1 = fminf(c1 + 0.5f * s1, 192.f), hi2 = fminf(c2 + 0.5f * s2, 192.f);
  float n0 = fmaxf(hi0 - lo0, 0.f), n1 = fmaxf(hi1 - lo1, 0.f), n2 = fmaxf(hi2 - lo2, 0.f);
  float vol = n0 * n1 * n2;
  float percent = vol / (s0 * s1 * s2);
  bool good = (percent > 0.1f) && (vol >= 15.f);
  bool keep = has && (vol > 0.f) && good;
  bool rej  = has && (vol > 0.f) && !good;
  float nc0 = lo0 + 0.5f * n0, nc1 = lo1 + 0.5f * n1, nc2 = lo2 + 0.5f * n2;
  float* p = prep + i * PREP_STRIDE;
  if (keep) {
    p[0] = nc0 * 0.25f; p[1] = nc1 * 0.25f; p[2] = nc2 * 0.25f;   // ctr/stride (stride=4)
    p[3] = 0.5f * n0;   p[4] = 0.5f * n1;   p[5] = 0.5f * n2;     // half sizes
    p[6] = nc0; p[7] = nc1; p[8] = nc2; p[9] = n0; p[10] = n1; p[11] = n2;
  } else {
    p[0] = -0.25f; p[1] = -0.25f; p[2] = -0.25f;   // (-1)/stride
    p[3] = -0.5f;  p[4] = -0.5f;  p[5] = -0.5f;    // (-1)/2
    p[6] = -1.f; p[7] = -1.f; p[8] = -1.f; p[9] = -1.f; p[10] = -1.f; p[11] = -1.f;
  }
  p[12] = lo0; p[13] = lo1; p[14] = lo2;
  p[15] = hi0; p[16] = hi1; p[17] = hi2;
  p[18] = keep ? 1.f : 0.f;
  p[19] = rej ? 1.f : 0.f;
}

// d2[(b*N+n)*A + a] = 4*(cz-z)^2 + (cy-y)^2 + (cx-x)^2 via WMMA cross term.
// A-tile 16x4 f32: rows = centers (n<8 valid, rest zero), K = (4*cz, cy, cx, 0)
// B-tile 4x16 f32: cols = anchors, K = (z, y, x, 0)
__global__ void k_dist_wmma(const float* __restrict__ prep, float* __restrict__ d2) {
  int blk = blockIdx.x;
  int b = blk / 432;          // 432 = A/128 chunks per image
  int chunk = blk % 432;
  int tid = threadIdx.x;
  int wave = tid >> 5, lane = tid & 31;
  int tile = chunk * 128 + wave * 16;
  int col = lane & 15;
  int hi = lane >> 4;         // 0: K=0,1 ; 1: K=2,3
  const float* pb = prep + (b * N_) * PREP_STRIDE;

  // A-matrix registers (branchless so EXEC stays all-ones through WMMA)
  int mm = col & 7;
  const float* pr = pb + mm * PREP_STRIDE;
  float vmask = (col < N_) ? 1.f : 0.f;
  float cz = pr[0] * vmask, cy = pr[1] * vmask, cx = pr[2] * vmask;
  float a0 = hi ? cx : 4.f * cz;
  float a1 = hi ? 0.f : cy;

  // B-matrix registers (anchor coords)
  int ai = tile + col;
  int z = ai / HW_;
  int rem = ai - z * HW_;
  int y = rem / W_;
  int x = rem - y * W_;
  float fz = (float)z, fy = (float)y, fx = (float)x;
  float b0 = hi ? fx : fz;
  float b1 = hi ? 0.f : fy;

  v2f av = {a0, a1};
  v2f bv = {b0, b1};
  v8f cv = {};
  cv = __builtin_amdgcn_wmma_f32_16x16x4_f32(
      /*neg_a=*/false, av, /*neg_b=*/false, bv,
      /*c_mod=*/(short)0, cv, /*reuse_a=*/false, /*reuse_b=*/false);

  if (hi == 0) {  // lanes 0..15 hold rows M=0..7 in vgprs 0..7
    float an = 4.f * fz * fz + fy * fy + fx * fx;
    #pragma unroll
    for (int v = 0; v < 8; ++v) {
      const float* pv = pb + v * PREP_STRIDE;
      float z0 = pv[0], y0 = pv[1], x0 = pv[2];
      float cn = 4.f * z0 * z0 + y0 * y0 + x0 * x0;
      d2[(size_t)(b * N_ + v) * A_ + ai] = cn - 2.f * cv[v] + an;
    }
  }
}

// Per (b,n): dual binary search for rank-7 and rank-189 cutoffs over 49-bit keys,
// then atomic scatter into pos_n (argmin n among top-7 owners) and ign.
__global__ void k_select(const float* __restrict__ prep, const float* __restrict__ d2,
                         int* __restrict__ pos_n, int* __restrict__ ign) {
  int idx = blockIdx.x;           // 0..127  (b*8+n)
  int b = idx >> 3, n = idx & 7;
  if (prep[idx * PREP_STRIDE + 18] == 0.f) return;  // mask_gt gate, block-uniform
  const float* row = d2 + (size_t)idx * A_;
  int tid = threadIdx.x;
  __shared__ int c7s, c9s;
  unsigned long long lo7 = 0, hi7 = (1ull << 49) - 1;
  unsigned long long lo9 = 0, hi9 = (1ull << 49) - 1;
  for (int it = 0; it < 49; ++it) {
    unsigned long long m7 = (lo7 + hi7) >> 1, m9 = (lo9 + hi9) >> 1;
    if (tid == 0) { c7s = 0; c9s = 0; }
    __syncthreads();
    int c7 = 0, c9 = 0;
    for (int a = tid; a < A_; a += 256) {
      unsigned long long key = ((unsigned long long)__float_as_uint(row[a]) << 17) | (unsigned)a;
      c7 += (key <= m7);
      c9 += (key <= m9);
    }
    atomicAdd(&c7s, c7);
    atomicAdd(&c9s, c9);
    __syncthreads();
    if (c7s >= 7)   hi7 = m7; else lo7 = m7 + 1;
    if (c9s >= 189) hi9 = m9; else lo9 = m9 + 1;
    __syncthreads();
  }
  unsigned long long K7 = hi7, K189 = hi9;
  for (int a = tid; a < A_; a += 256) {
    unsigned long long key = ((unsigned long long)__float_as_uint(row[a]) << 17) | (unsigned)a;
    if (key <= K7)        atomicMin(&pos_n[b * A_ + a], n);
    else if (key <= K189) atomicOr(&ign[b * A_ + a], 1);
  }
}

__global__ void k_per_anchor(const float* __restrict__ prep, const float* __restrict__ cls,
                             const float* __restrict__ shp, const float* __restrict__ off,
                             const int* __restrict__ pos_n, const int* __restrict__ ign,
                             float* __restrict__ neg, float* accum, int* npos) {
  constexpr int BPI = A_ / 256;   // 216
  int b = blockIdx.x / BPI;
  int a = (blockIdx.x % BPI) * 256 + threadIdx.x;
  __shared__ float sp[N_ * PREP_STRIDE];
  for (int i = threadIdx.x; i < N_ * PREP_STRIDE; i += 256)
    sp[i] = prep[b * N_ * PREP_STRIDE + i];
  __syncthreads();

  int z = a / HW_; int rem = a - z * HW_; int y = rem / W_; int x = rem - y * W_;
  float fz = (float)z, fy = (float)y, fx = (float)x;

  // grid ignore (reference compares voxel index against world coords — replicated)
  bool gi = false;
  for (int n = 0; n < N_; ++n) {
    const float* p = sp + n * PREP_STRIDE;
    if (p[19] != 0.f &&
        fz >= floorf(p[12]) && fz < ceilf(p[15]) &&
        fy >= floorf(p[13]) && fy < ceilf(p[16]) &&
        fx >= floorf(p[14]) && fx < ceilf(p[17])) gi = true;
  }

  int pn = pos_n[b * A_ + a];
  bool fg = pn < N_;
  int gt = fg ? pn : 0;
  float t = fg ? 1.f : 0.f;
  float ignore = (ign[b * A_ + a] ? -1.f : 0.f) + (gi ? -1.f : 0.f);

  float pred = cls[(size_t)b * A_ + a];
  float prob = 1.f / (1.f + expf(-pred));
  prob = fminf(fmaxf(prob, 1e-4f), 1.f - 1e-4f);
  float alpha_f = fg ? 0.75f : 0.25f;
  float pt = fg ? (1.f - prob) : prob;
  float fw = alpha_f * pt * pt;
  float bce = log1pf(expf(-fabsf(pred))) + fmaxf(pred, 0.f) - pred * t;
  float loss = (ignore == 0.f) ? fw * bce : 0.f;
  if (prob < 0.8f && fg) loss *= 4.f;
  neg[(size_t)b * A_ + a] = fg ? -1.f : loss;

  if (fg) {
    atomicAdd(&accum[b], loss);
    atomicAdd(&npos[b], 1);
    const float* pg = sp + gt * PREP_STRIDE;
    float to0 = pg[0] - fz, to1 = pg[1] - fy, to2 = pg[2] - fx;
    float po0 = off[((size_t)b * 3 + 0) * A_ + a];
    float po1 = off[((size_t)b * 3 + 1) * A_ + a];
    float po2 = off[((size_t)b * 3 + 2) * A_ + a];
    float ps0 = shp[((size_t)b * 3 + 0) * A_ + a];
    float ps1 = shp[((size_t)b * 3 + 1) * A_ + a];
    float ps2 = shp[((size_t)b * 3 + 2) * A_ + a];
    float offl = fabsf(po0 - to0) + fabsf(po1 - to1) + fabsf(po2 - to2);
    float shl  = fabsf(ps0 - pg[3]) + fabsf(ps1 - pg[4]) + fabsf(ps2 - pg[5]);
    // DIoU: pred box (world units) vs target bbox
    float pc0 = (fz + po0) * 4.f, pc1 = (fy + po1) * 4.f, pc2 = (fx + po2) * 4.f;
    float sz0 = 2.f * ps0, sz1 = 2.f * ps1, sz2 = 2.f * ps2;
    float tb0 = pg[6], tb1 = pg[7], tb2 = pg[8], tb3 = pg[9], tb4 = pg[10], tb5 = pg[11];
    float l10 = pc0 - 0.5f * sz0, h10 = pc0 + 0.5f * sz0;
    float l11 = pc1 - 0.5f * sz1, h11 = pc1 + 0.5f * sz1;
    float l12 = pc2 - 0.5f * sz2, h12 = pc2 + 0.5f * sz2;
    float l20 = tb0 - 0.5f * tb3, h20 = tb0 + 0.5f * tb3;
    float l21 = tb1 - 0.5f * tb4, h21 = tb1 + 0.5f * tb4;
    float l22 = tb2 - 0.5f * tb5, h22 = tb2 + 0.5f * tb5;
    float i0 = fmaxf(fminf(h10, h20) - fmaxf(l10, l20), 0.f);
    float i1 = fmaxf(fminf(h11, h21) - fmaxf(l11, l21), 0.f);
    float i2 = fmaxf(fminf(h12, h22) - fmaxf(l12, l22), 0.f);
    float inter = i0 * i1 * i2 + 1e-7f;
    float uni = sz0 * sz1 * sz2 + tb3 * tb4 * tb5 - inter;
    float iou = inter / uni;
    float e0 = fmaxf(h10, h20) - fminf(l10, l20);
    float e1 = fmaxf(h11, h21) - fminf(l11, l21);
    float e2 = fmaxf(h12, h22) - fminf(l12, l22);
    float c2d = e0 * e0 + e1 * e1 + e2 * e2 + 1e-7f;
    float r0 = (l20 + h20) - (l10 + h10);
    float r1 = (l21 + h21) - (l11 + h11);
    float r2 = (l22 + h22) - (l12 + h12);
    float rho2 = (r0 * r0 + r1 * r1 + r2 * r2) * 0.25f;
    float diou = iou - rho2 / c2d;
    atomicAdd(&accum[17], shl);
    atomicAdd(&accum[18], offl);
    atomicAdd(&accum[19], diou);
    atomicAdd(&accum[20], 1.f);
  }
}

// Per image: exact sum of k largest non-negative neg losses via bitwise kth-largest.
__global__ void k_hardneg(const float* __restrict__ neg, const int* __restrict__ npos, float* accum) {
  int b = blockIdx.x;
  const float* v = neg + (size_t)b * A_;
  int np = npos[b];
  int k = np > 0 ? 100 * np : 100;
  int tid = threadIdx.x;
  __shared__ int sc;
  __shared__ float ssum;
  unsigned X = 0;
  for (int bit = 30; bit >= 0; --bit) {
    unsigned C = X | (1u << bit);
    if (tid == 0) sc = 0;
    __syncthreads();
    int c = 0;
    for (int a = tid; a < A_; a += 256) {
      float f = v[a];
      if (f >= 0.f && __float_as_uint(f) >= C) c++;
    }
    atomicAdd(&sc, c);
    __syncthreads();
    if (sc >= k) X = C;
    __syncthreads();
  }
  if (tid == 0) { sc = 0; ssum = 0.f; }
  __syncthreads();
  int m = 0; float s = 0.f;
  for (int a = tid; a < A_; a += 256) {
    float f = v[a];
    if (f >= 0.f && __float_as_uint(f) > X) { m++; s += f; }
  }
  atomicAdd(&sc, m);
  atomicAdd(&ssum, s);
  __syncthreads();
  if (tid == 0) {
    float negsum = ssum + (float)(k - sc) * __uint_as_float(X);
    float per = (accum[b] + negsum) / fmaxf((float)np, 1.f);
    atomicAdd(&accum[16], per * (1.f / 16.f));
  }
}

__global__ void k_final(const float* __restrict__ accum, float* __restrict__ out) {
  if (threadIdx.x == 0 && blockIdx.x == 0) {
    float denom = fmaxf(accum[20], 1.f);
    out[0] = accum[16];
    out[1] = accum[17] / (denom * 3.f);
    out[2] = accum[18] / (denom * 3.f);
    out[3] = 1.f - accum[19] / denom;
  }
}

extern "C" void kernel_launch(void* const* d_in, const int* in_sizes, int n_in,
                              void* d_out, int out_size, void* d_ws, size_t ws_size,
                              hipStream_t stream) {
  const float* cls = (const float*)d_in[0];
  const float* shp = (const float*)d_in[1];
  const float* off = (const float*)d_in[2];
  const float* ann = (const float*)d_in[3];
  float* out = (float*)d_out;

  float* prep  = (float*)d_ws;                 // 128*20 floats
  float* accum = prep + BN_ * PREP_STRIDE;     // 24 floats
  int*   npos  = (int*)(accum + 24);           // 32 ints (16 used)
  int*   pos_n = npos + 32;                    // BA ints
  int*   ign   = pos_n + BA_;                  // BA ints
  float* neg   = (float*)(ign + BA_);          // BA floats
  float* d2    = neg + BA_;                    // B*N*A floats (~28 MB, L2-resident)

  k_init<<<(BA_ + 255) / 256, 256, 0, stream>>>(pos_n, ign, accum, npos);
  k_prep<<<1, 128, 0, stream>>>(ann, prep);
  k_dist_wmma<<<B_ * 432, 256, 0, stream>>>(prep, d2);
  k_select<<<BN_, 256, 0, stream>>>(prep, d2, pos_n, ign);
  k_per_anchor<<<B_ * (A_ / 256), 256, 0, stream>>>(prep, cls, shp, off, pos_n, ign, neg, accum, npos);
  k_hardneg<<<B_, 256, 0, stream>>>(neg, npos, accum);
  k_final<<<1, 1, 0, stream>>>(accum, out);
}